// GNN_38319698215421
// MI455X (gfx1250) — compile-verified
//
#include <hip/hip_runtime.h>
#include <hip/hip_bf16.h>

#define NN   50000
#define EE   800000
#define ETOT (EE + NN)   // edges + self loops
#define NEG_SLOPE 0.2f

typedef float v2f __attribute__((ext_vector_type(2)));
typedef float v8f __attribute__((ext_vector_type(8)));

// ---------- monotone unsigned key for float atomic-max ----------
__device__ __forceinline__ unsigned fkey(float f) {
    unsigned u = __float_as_uint(f);
    return (u & 0x80000000u) ? ~u : (u | 0x80000000u);
}
__device__ __forceinline__ float fdecode(unsigned k) {
    unsigned u = (k & 0x80000000u) ? (k & 0x7FFFFFFFu) : ~k;
    return __uint_as_float(u);
}
__device__ __forceinline__ float lrelu(float v) {
    return (v > 0.0f) ? v : NEG_SLOPE * v;
}

// ---------- Layer 1: h = [x|x_pos] @ W1 (rank-2), fused attention dots ----------
__global__ void layer1_kernel(const float* __restrict__ x, const float* __restrict__ xpos,
                              const float* __restrict__ W1, const float* __restrict__ as,
                              const float* __restrict__ ad,
                              float* __restrict__ h, float* __restrict__ a_src,
                              float* __restrict__ a_dst) {
    int n = blockIdx.x;          // one block per node, 256 threads
    int j = threadIdx.x;         // feature index (H=2, C=128 -> F=256)
    float xv = x[n], pv = xpos[n];
    float hv = xv * W1[j] + pv * W1[256 + j];
    h[(size_t)n * 256 + j] = hv;
    __shared__ float rs[256], rd[256];
    rs[j] = hv * as[j];
    rd[j] = hv * ad[j];
    __syncthreads();
    int c = j & 127;
    for (int off = 64; off > 0; off >>= 1) {
        if (c < off) { rs[j] += rs[j + off]; rd[j] += rd[j + off]; }
        __syncthreads();
    }
    if (c == 0) {
        int head = j >> 7;
        a_src[n * 2 + head] = rs[j];
        a_dst[n * 2 + head] = rd[j];
    }
}

// ---------- fp32 WMMA GEMM: C[M,Nout] = A[M,K] * B[K,Nout], M%16==0, K%4==0 ----------
__global__ void gemm_wmma_f32(const float* __restrict__ A, const float* __restrict__ B,
                              float* __restrict__ Cout, int K, int Nout) {
    int wave = threadIdx.x >> 5;     // 8 waves per block
    int lane = threadIdx.x & 31;
    int row0 = blockIdx.x << 4;      // 16 rows per block
    int ntiles = Nout >> 4;
    int l15  = lane & 15;
    int khalf = (lane >> 4) << 1;    // lanes 0-15: K pair (k0,k0+1); 16-31: (k0+2,k0+3)
    const float* Arow = A + (size_t)(row0 + l15) * K;
    for (int nt = wave; nt < ntiles; nt += 8) {
        int col0 = nt << 4;
        v8f acc = {};
        for (int k0 = 0; k0 < K; k0 += 4) {
            int kk = k0 + khalf;
            v2f a; a.x = Arow[kk];                      a.y = Arow[kk + 1];
            v2f b; b.x = B[(size_t)kk * Nout + col0 + l15];
                   b.y = B[(size_t)(kk + 1) * Nout + col0 + l15];
            acc = __builtin_amdgcn_wmma_f32_16x16x4_f32(
                false, a, false, b, (short)0, acc, false, false);
        }
        int rowoff = (lane < 16) ? 0 : 8;
        float* Crow = Cout + (size_t)(row0 + rowoff) * Nout + col0 + l15;
#pragma unroll
        for (int v = 0; v < 8; ++v) Crow[(size_t)v * Nout] = acc[v];
    }
}

// ---------- per-node attention dots: a_src/a_dst[n,h] = sum_c h[n,h,c]*att[h,c] ----------
__global__ void node_attn_kernel(const float* __restrict__ h, const float* __restrict__ as,
                                 const float* __restrict__ ad, float* __restrict__ a_src,
                                 float* __restrict__ a_dst, int F, int H) {
    int n = blockIdx.x;
    int j = threadIdx.x;            // blockDim == F (128 or 256), C == 128
    float hv = h[(size_t)n * F + j];
    __shared__ float rs[256], rd[256];
    rs[j] = hv * as[j];
    rd[j] = hv * ad[j];
    __syncthreads();
    int c = j & 127;
    for (int off = 64; off > 0; off >>= 1) {
        if (c < off) { rs[j] += rs[j + off]; rd[j] += rd[j + off]; }
        __syncthreads();
    }
    if (c == 0) {
        int head = j >> 7;
        a_src[n * H + head] = rs[j];
        a_dst[n * H + head] = rd[j];
    }
}

// ---------- clear per-layer scratch ----------
__global__ void init_layer_kernel(unsigned* __restrict__ mk, float* __restrict__ ss,
                                  float* __restrict__ ac, int nms, int nac) {
    int i = blockIdx.x * blockDim.x + threadIdx.x;
    if (i < nms) { mk[i] = 0u; ss[i] = 0.0f; }    // key 0 == minimal unsigned -> any edge wins
    if (i < nac) ac[i] = 0.0f;
}

// ---------- pass 1: segment max via atomic max on ordered keys ----------
__global__ void edge_max_kernel(const int* __restrict__ eidx, const float* __restrict__ a_src,
                                const float* __restrict__ a_dst, unsigned* __restrict__ mk,
                                int H) {
    int t = blockIdx.x * blockDim.x + threadIdx.x;
    int e = t / H, h = t - e * H;
    if (e >= ETOT) return;
    int s = (e < EE) ? eidx[e]      : (e - EE);
    int d = (e < EE) ? eidx[EE + e] : (e - EE);
    float v = lrelu(a_src[s * H + h] + a_dst[d * H + h]);
    atomicMax(&mk[d * H + h], fkey(v));
}

// ---------- pass 2: denominator + unnormalized numerator scatter-add ----------
__global__ void edge_aggr_kernel(const int* __restrict__ eidx, const float* __restrict__ a_src,
                                 const float* __restrict__ a_dst, const unsigned* __restrict__ mk,
                                 float* __restrict__ ssum, const float* __restrict__ hin,
                                 float* __restrict__ accum, int F, int H) {
    int wave = threadIdx.x >> 5;
    int lane = threadIdx.x & 31;
    int e = blockIdx.x * 8 + wave;           // one wave per edge
    if (e >= ETOT) return;
    int s = (e < EE) ? eidx[e]      : (e - EE);
    int d = (e < EE) ? eidx[EE + e] : (e - EE);
    int C = F / H;
    float ex[2];
    for (int h = 0; h < H; ++h) {
        float v = lrelu(a_src[s * H + h] + a_dst[d * H + h]);
        float m = fdecode(mk[d * H + h]);
        ex[h] = __expf(v - m);
        if (lane == 0) atomicAdd(&ssum[d * H + h], ex[h]);
    }
    const float* hs = hin + (size_t)s * F;
    float* ac = accum + (size_t)d * F;
    for (int j = lane; j < F; j += 32)
        atomicAdd(&ac[j], ex[j / C] * hs[j]);
}

// ---------- normalize by softmax denominator + bias ----------
__global__ void normalize_kernel(const float* __restrict__ accum, const float* __restrict__ ssum,
                                 const float* __restrict__ bias, float* __restrict__ hout,
                                 int F, int H) {
    size_t i = (size_t)blockIdx.x * blockDim.x + threadIdx.x;
    if (i >= (size_t)NN * F) return;
    int n = (int)(i / F);
    int j = (int)(i - (size_t)n * F);
    int C = F / H;
    hout[i] = accum[i] / ssum[n * H + j / C] + bias[j];
}

// ---------- final mean over nodes ----------
__global__ void zero_out_kernel(float* __restrict__ out) {
    if (threadIdx.x < 128) out[threadIdx.x] = 0.0f;
}
__global__ void col_mean_kernel(const float* __restrict__ h3, float* __restrict__ out) {
    int j = threadIdx.x;   // 128 threads
    float acc = 0.0f;
    for (int n = blockIdx.x; n < NN; n += gridDim.x)
        acc += h3[(size_t)n * 128 + j];
    atomicAdd(&out[j], acc * (1.0f / (float)NN));
}

extern "C" void kernel_launch(void* const* d_in, const int* in_sizes, int n_in,
                              void* d_out, int out_size, void* d_ws, size_t ws_size,
                              hipStream_t stream) {
    const float* x    = (const float*)d_in[0];
    const int*   eidx = (const int*)  d_in[1];
    const float* xpos = (const float*)d_in[2];
    const float* W1   = (const float*)d_in[3];
    const float* as1  = (const float*)d_in[4];
    const float* ad1  = (const float*)d_in[5];
    const float* b1   = (const float*)d_in[6];
    const float* W2   = (const float*)d_in[7];
    const float* as2  = (const float*)d_in[8];
    const float* ad2  = (const float*)d_in[9];
    const float* b2   = (const float*)d_in[10];
    const float* W3   = (const float*)d_in[11];
    const float* as3  = (const float*)d_in[12];
    const float* ad3  = (const float*)d_in[13];
    const float* b3   = (const float*)d_in[14];

    float*    bufA  = (float*)d_ws;                        // GEMM / layer-1 output [N,256]
    float*    bufB  = bufA + (size_t)NN * 256;             // scatter accumulator   [N,256]
    float*    bufC  = bufB + (size_t)NN * 256;             // normalized layer out  [N,256]
    float*    a_src = bufC + (size_t)NN * 256;             // [N,2]
    float*    a_dst = a_src + (size_t)NN * 2;              // [N,2]
    unsigned* mkeys = (unsigned*)(a_dst + (size_t)NN * 2); // [N,2]
    float*    ssum  = (float*)(mkeys + (size_t)NN * 2);    // [N,2]

    const int TB = 256;
    int gInit256 = (NN * 256 + TB - 1) / TB;
    int gInit128 = (NN * 128 + TB - 1) / TB;
    int gEdgeH2  = (ETOT * 2 + TB - 1) / TB;
    int gEdgeH1  = (ETOT * 1 + TB - 1) / TB;
    int gAggr    = (ETOT + 7) / 8;
    int gGemm    = NN / 16;   // 3125, exact

    // ---- Layer 1: F=256, H=2 ----
    layer1_kernel<<<NN, 256, 0, stream>>>(x, xpos, W1, as1, ad1, bufA, a_src, a_dst);
    init_layer_kernel<<<gInit256, TB, 0, stream>>>(mkeys, ssum, bufB, NN * 2, NN * 256);
    edge_max_kernel<<<gEdgeH2, TB, 0, stream>>>(eidx, a_src, a_dst, mkeys, 2);
    edge_aggr_kernel<<<gAggr, TB, 0, stream>>>(eidx, a_src, a_dst, mkeys, ssum, bufA, bufB, 256, 2);
    normalize_kernel<<<gInit256, TB, 0, stream>>>(bufB, ssum, b1, bufC, 256, 2);

    // ---- Layer 2: F=256, H=2 ----
    gemm_wmma_f32<<<gGemm, 256, 0, stream>>>(bufC, W2, bufA, 256, 256);
    node_attn_kernel<<<NN, 256, 0, stream>>>(bufA, as2, ad2, a_src, a_dst, 256, 2);
    init_layer_kernel<<<gInit256, TB, 0, stream>>>(mkeys, ssum, bufB, NN * 2, NN * 256);
    edge_max_kernel<<<gEdgeH2, TB, 0, stream>>>(eidx, a_src, a_dst, mkeys, 2);
    edge_aggr_kernel<<<gAggr, TB, 0, stream>>>(eidx, a_src, a_dst, mkeys, ssum, bufA, bufB, 256, 2);
    normalize_kernel<<<gInit256, TB, 0, stream>>>(bufB, ssum, b2, bufC, 256, 2);

    // ---- Layer 3: F=128, H=1 ----
    gemm_wmma_f32<<<gGemm, 256, 0, stream>>>(bufC, W3, bufA, 256, 128);
    node_attn_kernel<<<NN, 128, 0, stream>>>(bufA, as3, ad3, a_src, a_dst, 128, 1);
    init_layer_kernel<<<gInit128, TB, 0, stream>>>(mkeys, ssum, bufB, NN * 1, NN * 128);
    edge_max_kernel<<<gEdgeH1, TB, 0, stream>>>(eidx, a_src, a_dst, mkeys, 1);
    edge_aggr_kernel<<<gAggr, TB, 0, stream>>>(eidx, a_src, a_dst, mkeys, ssum, bufA, bufB, 128, 1);
    normalize_kernel<<<gInit128, TB, 0, stream>>>(bufB, ssum, b3, bufC, 128, 1);

    // ---- mean over nodes -> [128] ----
    zero_out_kernel<<<1, 128, 0, stream>>>((float*)d_out);
    col_mean_kernel<<<256, 128, 0, stream>>>(bufC, (float*)d_out);
}